// NNBase_61838939128123
// MI455X (gfx1250) — compile-verified
//
#include <hip/hip_runtime.h>
#include <math.h>

typedef float v2f __attribute__((ext_vector_type(2)));
typedef float v8f __attribute__((ext_vector_type(8)));

#define DD 64
#define BB 64
#define NG 256   // 4*D gate width

// ---------------------------------------------------------------------------
// WMMA helper: acc += A(16xK, row-major lda) * W^T, where W is (ncols x K)
// row-major with leading dim ldw; i.e. B[k][n] = W[n*ldw + k].
// Uses V_WMMA_F32_16X16X4_F32 (fp32 in/out — full precision).
// A 16x4 f32 layout: lanes 0-15 rows M=0-15 {K=k0,k0+1}, lanes 16-31 {k0+2,k0+3}
// B 4x16 f32 layout: mirrored on N. C/D: VGPR j -> M=j (+8 for upper half-wave).
// ---------------------------------------------------------------------------
__device__ __forceinline__ v8f wmma_tile_f32(const float* __restrict__ A, int lda,
                                             const float* __restrict__ W, int ldw,
                                             int K, v8f acc) {
  const int lane = threadIdx.x & 31;
  const int half = lane >> 4;
  const int idx  = lane & 15;
  for (int k0 = 0; k0 < K; k0 += 4) {
    const int k = k0 + 2 * half;
    v2f a, b;
    a.x = A[idx * lda + k];
    a.y = A[idx * lda + k + 1];
    b.x = W[idx * ldw + k];
    b.y = W[idx * ldw + k + 1];
    acc = __builtin_amdgcn_wmma_f32_16x16x4_f32(false, a, false, b, (short)0, acc,
                                                false, false);
  }
  return acc;
}

__device__ __forceinline__ float sigmoidf(float x) { return 1.0f / (1.0f + __expf(-x)); }

// Monotone float <-> uint mapping for atomic float-max via atomicMax(u32).
__device__ __forceinline__ unsigned fkey(float f) {
  unsigned b = __float_as_uint(f);
  return (b & 0x80000000u) ? ~b : (b | 0x80000000u);
}
__device__ __forceinline__ float funkey(unsigned k) {
  return __uint_as_float((k & 0x80000000u) ? (k ^ 0x80000000u) : ~k);
}

// ---------------------------------------------------------------------------
__global__ void zero_kernel(unsigned* __restrict__ p, int n) {
  int i = blockIdx.x * blockDim.x + threadIdx.x;
  if (i < n) p[i] = 0u;
}

// Scatter-sum edge features onto dst nodes + degree count (bandwidth-bound).
__global__ void edge_scatter_kernel(const float* __restrict__ efeat,
                                    const int* __restrict__ edge_dst,
                                    float* __restrict__ edge_agg,
                                    float* __restrict__ deg, int total) {
  int i = blockIdx.x * blockDim.x + threadIdx.x;
  if (i >= total) return;
  int e = i >> 6, d = i & 63;
  int dst = edge_dst[e];
  atomicAdd(&edge_agg[dst * DD + d], efeat[i]);
  if (d == 0) atomicAdd(&deg[dst], 1.0f);
}

__global__ void edge_mean_kernel(float* __restrict__ edge_agg,
                                 const float* __restrict__ deg, int total) {
  int i = blockIdx.x * blockDim.x + threadIdx.x;
  if (i >= total) return;
  edge_agg[i] /= fmaxf(deg[i >> 6], 1.0f);
}

// ---------------------------------------------------------------------------
// LSTM cell: gates = q_star @ Wih^T + h @ Whh^T + (bih+bhh), gate order i,f,g,o
// One block, 8 waves; 64 16x16 output tiles of the (64 x 256) gate matrix,
// each tile accumulated with v_wmma_f32_16x16x4_f32.
// ---------------------------------------------------------------------------
__global__ void lstm_kernel(const float* __restrict__ Wih,   // (256,128)
                            const float* __restrict__ Whh,   // (256,64)
                            const float* __restrict__ bih,   // (256)
                            const float* __restrict__ bhh,   // (256)
                            const float* __restrict__ q_star,// (64,128)
                            float* __restrict__ h,           // (64,64)
                            float* __restrict__ c) {         // (64,64)
  __shared__ float gates[BB * NG];   // 64 KB
  const int wave = threadIdx.x >> 5;
  const int lane = threadIdx.x & 31;
  const int half = lane >> 4;
  const int idx  = lane & 15;
  for (int t = wave; t < 64; t += 8) {
    const int mt = t & 3, nt = t >> 2;
    v8f acc = {};
    acc = wmma_tile_f32(q_star + mt * 16 * 128, 128, Wih + nt * 16 * 128, 128, 128, acc);
    acc = wmma_tile_f32(h      + mt * 16 * DD,  DD,  Whh + nt * 16 * DD,  DD,  DD,  acc);
    const int n = nt * 16 + idx;
    const float bn = bih[n] + bhh[n];
#pragma unroll
    for (int j = 0; j < 8; ++j)
      gates[(mt * 16 + j + 8 * half) * NG + n] = acc[j] + bn;
  }
  __syncthreads();
  for (int i = threadIdx.x; i < BB * DD; i += blockDim.x) {
    const int b = i >> 6, d = i & 63;
    const float ig = sigmoidf(gates[b * NG + d]);
    const float fg = sigmoidf(gates[b * NG + 64 + d]);
    const float gg = tanhf(gates[b * NG + 128 + d]);
    const float og = sigmoidf(gates[b * NG + 192 + d]);
    const float cn = fg * c[i] + ig * gg;
    c[i] = cn;
    h[i] = og * tanhf(cn);
  }
}

// e[n] = dot(X[n,:], h[gid[n],:]); segment max via ordered-uint atomicMax.
// One wave (32 lanes) per node; each lane covers 2 of the 64 features.
__global__ void attn_logits_kernel(const float* __restrict__ X,
                                   const float* __restrict__ h,
                                   const int* __restrict__ gid,
                                   float* __restrict__ e,
                                   unsigned* __restrict__ mbuf, int N) {
  const int node = blockIdx.x * (blockDim.x >> 5) + (threadIdx.x >> 5);
  if (node >= N) return;
  const int lane = threadIdx.x & 31;
  const int g = gid[node];
  float v = X[node * DD + lane]      * h[g * DD + lane] +
            X[node * DD + 32 + lane] * h[g * DD + 32 + lane];
#pragma unroll
  for (int off = 16; off; off >>= 1) v += __shfl_xor(v, off, 32);
  if (lane == 0) {
    e[node] = v;
    atomicMax(&mbuf[g], fkey(v));
  }
}

__global__ void attn_exp_kernel(const float* __restrict__ e,
                                const unsigned* __restrict__ mbuf,
                                const int* __restrict__ gid,
                                float* __restrict__ ex,
                                float* __restrict__ sbuf, int N) {
  int n = blockIdx.x * blockDim.x + threadIdx.x;
  if (n >= N) return;
  const int g = gid[n];
  const float v = __expf(e[n] - funkey(mbuf[g]));
  ex[n] = v;
  atomicAdd(&sbuf[g], v);
}

__global__ void attn_readout_kernel(const float* __restrict__ X,
                                    const float* __restrict__ ex,
                                    const float* __restrict__ sbuf,
                                    const int* __restrict__ gid,
                                    float* __restrict__ readout, int total) {
  int i = blockIdx.x * blockDim.x + threadIdx.x;
  if (i >= total) return;
  const int n = i >> 6, d = i & 63;
  const int g = gid[n];
  const float alpha = ex[n] / sbuf[g];
  atomicAdd(&readout[g * DD + d], X[i] * alpha);
}

__global__ void qstar_kernel(const float* __restrict__ h,
                             const float* __restrict__ readout,
                             float* __restrict__ q_star) {
  int i = blockIdx.x * blockDim.x + threadIdx.x;
  if (i >= BB * 2 * DD) return;
  const int b = i >> 7, j = i & 127;
  q_star[i] = (j < DD) ? h[b * DD + j] : readout[b * DD + (j - DD)];
}

// ---------------------------------------------------------------------------
// Final MLP: relu([qs_node|qs_edge] @ W1^T + b1) -> relu(@W2^T+b2) -> @W3^T+b3
// Layers 1&2 via v_wmma_f32_16x16x4_f32; layer 3 is a 16-wide dot per row.
// ---------------------------------------------------------------------------
__global__ void mlp_kernel(const float* __restrict__ qs_node,  // (64,128)
                           const float* __restrict__ qs_edge,  // (64,128)
                           const float* __restrict__ W1, const float* __restrict__ b1,
                           const float* __restrict__ W2, const float* __restrict__ b2,
                           const float* __restrict__ W3, const float* __restrict__ b3,
                           float* __restrict__ out) {          // (64)
  __shared__ float y1[BB * 32];
  __shared__ float y2[BB * 16];
  const int wave = threadIdx.x >> 5;
  const int lane = threadIdx.x & 31;
  const int half = lane >> 4;
  const int idx  = lane & 15;
  // Layer 1: (64x256)@(256x32): 4 M-tiles x 2 N-tiles = 8 tiles, one per wave.
  {
    const int mt = wave & 3, nt = wave >> 2;
    v8f acc = {};
    // K=0..127 from qs_node, K=128..255 from qs_edge (concat along K).
    acc = wmma_tile_f32(qs_node + mt * 16 * 128, 128, W1 + nt * 16 * 256,       256, 128, acc);
    acc = wmma_tile_f32(qs_edge + mt * 16 * 128, 128, W1 + nt * 16 * 256 + 128, 256, 128, acc);
    const int n = nt * 16 + idx;
    const float bn = b1[n];
#pragma unroll
    for (int j = 0; j < 8; ++j)
      y1[(mt * 16 + j + 8 * half) * 32 + n] = fmaxf(acc[j] + bn, 0.0f);
  }
  __syncthreads();
  // Layer 2: (64x32)@(32x16): 4 M-tiles, single 16-wide N tile.
  if (wave < 4) {
    v8f acc = {};
    acc = wmma_tile_f32(y1 + wave * 16 * 32, 32, W2, 32, 32, acc);
    const float bn = b2[idx];
#pragma unroll
    for (int j = 0; j < 8; ++j)
      y2[(wave * 16 + j + 8 * half) * 16 + idx] = fmaxf(acc[j] + bn, 0.0f);
  }
  __syncthreads();
  // Layer 3: out[b] = y2[b,:16] . W3 + b3
  if (threadIdx.x < BB) {
    float s = b3[0];
#pragma unroll
    for (int k = 0; k < 16; ++k) s += y2[threadIdx.x * 16 + k] * W3[k];
    out[threadIdx.x] = s;
  }
}

// ---------------------------------------------------------------------------
static void run_set2set(const float* X, int N, const int* gid,
                        const float* Wih, const float* Whh,
                        const float* bih, const float* bhh,
                        float* q_star, float* h, float* c,
                        float* e, float* ex,
                        unsigned* mbuf, float* sbuf, float* readout,
                        hipStream_t stream) {
  for (int it = 0; it < 3; ++it) {
    lstm_kernel<<<1, 256, 0, stream>>>(Wih, Whh, bih, bhh, q_star, h, c);
    // mbuf | sbuf | readout are contiguous: 64 + 64 + 4096 = 4224 dwords
    zero_kernel<<<(4224 + 255) / 256, 256, 0, stream>>>(mbuf, 4224);
    attn_logits_kernel<<<(N + 7) / 8, 256, 0, stream>>>(X, h, gid, e, mbuf, N);
    attn_exp_kernel<<<(N + 255) / 256, 256, 0, stream>>>(e, mbuf, gid, ex, sbuf, N);
    attn_readout_kernel<<<(N * DD + 255) / 256, 256, 0, stream>>>(X, ex, sbuf, gid,
                                                                  readout, N * DD);
    qstar_kernel<<<(BB * 2 * DD + 255) / 256, 256, 0, stream>>>(h, readout, q_star);
  }
}

extern "C" void kernel_launch(void* const* d_in, const int* in_sizes, int n_in,
                              void* d_out, int out_size, void* d_ws, size_t ws_size,
                              hipStream_t stream) {
  const float* feat     = (const float*)d_in[0];
  const float* efeat    = (const float*)d_in[1];
  const int*   edge_dst = (const int*)d_in[2];
  const int*   gid      = (const int*)d_in[3];
  const float* node_Wih = (const float*)d_in[5];
  const float* node_Whh = (const float*)d_in[6];
  const float* node_bih = (const float*)d_in[7];
  const float* node_bhh = (const float*)d_in[8];
  const float* edge_Wih = (const float*)d_in[9];
  const float* edge_Whh = (const float*)d_in[10];
  const float* edge_bih = (const float*)d_in[11];
  const float* edge_bhh = (const float*)d_in[12];
  const float* W1 = (const float*)d_in[13];
  const float* b1 = (const float*)d_in[14];
  const float* W2 = (const float*)d_in[15];
  const float* b2 = (const float*)d_in[16];
  const float* W3 = (const float*)d_in[17];
  const float* b3 = (const float*)d_in[18];
  float* out = (float*)d_out;

  const int N = in_sizes[0] / DD;
  const int E = in_sizes[1] / DD;

  // Workspace layout (floats)
  float* ws = (float*)d_ws;
  size_t off = 0;
  float* edge_agg = ws + off; off += (size_t)N * DD;   // followed directly by deg
  float* deg      = ws + off; off += N;
  float* e        = ws + off; off += N;
  float* ex       = ws + off; off += N;
  float* qs_node  = ws + off; off += BB * 2 * DD;      // init1 zero starts here
  float* h        = ws + off; off += BB * DD;          // init2 zero starts here
  float* c        = ws + off; off += BB * DD;
  float* qs_edge  = ws + off; off += BB * 2 * DD;
  unsigned* mbuf  = (unsigned*)(ws + off); off += BB;  // reset zero starts here
  float* sbuf     = ws + off; off += BB;
  float* readout  = ws + off; off += BB * DD;

  // 1) edge mean-aggregation (dominant memory traffic: ~256 MB)
  const int agg_total = N * (DD + 1);                  // edge_agg + deg contiguous
  zero_kernel<<<(agg_total + 255) / 256, 256, 0, stream>>>((unsigned*)edge_agg, agg_total);
  edge_scatter_kernel<<<(E * DD + 255) / 256, 256, 0, stream>>>(efeat, edge_dst,
                                                                edge_agg, deg, E * DD);
  edge_mean_kernel<<<(N * DD + 255) / 256, 256, 0, stream>>>(edge_agg, deg, N * DD);

  // 2) Set2Set over node features (zero qs_node..readout in one shot)
  const int init1 = BB * 2 * DD + 2 * BB * DD + BB * 2 * DD + BB + BB + BB * DD;
  zero_kernel<<<(init1 + 255) / 256, 256, 0, stream>>>((unsigned*)qs_node, init1);
  run_set2set(feat, N, gid, node_Wih, node_Whh, node_bih, node_bhh,
              qs_node, h, c, e, ex, mbuf, sbuf, readout, stream);

  // 3) Set2Set over aggregated edge features (zero h..readout; keep qs_node)
  const int init2 = 2 * BB * DD + BB * 2 * DD + BB + BB + BB * DD;
  zero_kernel<<<(init2 + 255) / 256, 256, 0, stream>>>((unsigned*)h, init2);
  run_set2set(edge_agg, N, gid, edge_Wih, edge_Whh, edge_bih, edge_bhh,
              qs_edge, h, c, e, ex, mbuf, sbuf, readout, stream);

  // 4) Final MLP -> (64,1) output
  mlp_kernel<<<1, 256, 0, stream>>>(qs_node, qs_edge, W1, b1, W2, b2, W3, b3, out);
}